// BlackholeEmbeddings_6957847020093
// MI455X (gfx1250) — compile-verified
//
#include <hip/hip_runtime.h>
#include <hip/hip_bf16.h>

// ---------------------------------------------------------------------------
// BlackholeEmbeddings for MI455X (gfx1250, wave32).
// Bandwidth-bound embedding fusion + rare-path numeric MLP on fp32 WMMA.
// ---------------------------------------------------------------------------

#define HID   1024
#define SEQ   2048
#define NTOK  16384          // 8 * 2048
#define NF    94
#define FP    100            // feats LDS row stride (pad 94->100, conflict-free)
#define PI    256
#define H1S   260            // h1 LDS row stride
#define NS    1036           // num LDS row stride (16B-aligned rows, bank-spread)
#define NUM_ID 5
#define LN_EPS 1e-12f

typedef __attribute__((ext_vector_type(2))) float v2f;
typedef __attribute__((ext_vector_type(8))) float v8f;

__device__ __forceinline__ float wave_sum32(float v) {
#pragma unroll
  for (int m = 16; m > 0; m >>= 1) v += __shfl_xor(v, m, 32);
  return v;
}

// 94 numeric features, exact replica of the reference (fp64 internally).
__device__ void compute_feats(float valf, int fmt, float* __restrict__ f) {
  double v = (double)valf;
  long long bits = __double_as_longlong(v);
#pragma unroll
  for (int i = 0; i < 63; ++i) f[i] = ((bits >> i) & 1LL) ? 1.0f : 0.0f;
  f[63] = 0.0f;  // torch (1<<63) overflows to INT64_MIN; (x & m) > 0 always false
  double av = fabs(v);
  double fl = floor(av);
  int units = (int)fmin(fmax(fmod(fl, 10.0), 0.0), 9.0);
  int tens  = (int)fmin(fmax(fmod(floor(fl / 10.0), 10.0), 0.0), 9.0);
#pragma unroll
  for (int j = 0; j < 10; ++j) f[64 + j] = (units == j) ? 1.0f : 0.0f;
#pragma unroll
  for (int j = 0; j < 10; ++j) f[74 + j] = (tens == j) ? 1.0f : 0.0f;
  f[84] = (float)log(av + 1e-6);
  f[85] = (v > 0.0) ? 1.0f : ((v < 0.0) ? -1.0f : 0.0f);
  f[86] = (av > 1e-6) ? (float)floor(log10(fmax(av, 1e-300))) : 0.0f;
  f[87] = (av == fl) ? 1.0f : 0.0f;
  f[88] = (v > 0.0) ? 1.0f : 0.0f;
  f[89] = (v == 0.0) ? 1.0f : 0.0f;
  f[90] = (v < 0.0) ? 1.0f : 0.0f;
  bool pos_int = (v == floor(v)) && (v > 0.0);
  double l2 = log2(fmax(v, 1.0));
  f[91] = (pos_int && (l2 == floor(l2))) ? 1.0f : 0.0f;
  f[92] = (fmt == 0) ? 1.0f : 0.0f;
  f[93] = (fmt == 1) ? 1.0f : 0.0f;
#pragma unroll
  for (int i = NF; i < FP; ++i) f[i] = 0.0f;   // zero K-padding (rows 94..99)
}

__global__ __launch_bounds__(256) void blackhole_emb_kernel(
    const int* __restrict__ ids, const float* __restrict__ vals,
    const int* __restrict__ fmts,
    const float* __restrict__ Wword, const float* __restrict__ Wpos,
    const float* __restrict__ Wtype,
    const float* __restrict__ lng, const float* __restrict__ lnb,
    const float* __restrict__ pw1, const float* __restrict__ pb1,
    const float* __restrict__ pw2, const float* __restrict__ pb2,
    const float* __restrict__ plng, const float* __restrict__ plnb,
    float* __restrict__ out)
{
  extern __shared__ char smem_raw[];
  float* sFeats = (float*)smem_raw;        // 16 * FP
  float* sH1    = sFeats + 16 * FP;        // 16 * H1S
  float* sNum   = sH1 + 16 * H1S;          // 16 * NS  (pre-LN layer-2 output)
  int*   sAct   = (int*)(sNum + 16 * NS);  // 16

  const int tile = blockIdx.x;             // one tile = 16 tokens
  const int tid  = threadIdx.x;
  const int lane = tid & 31;
  const int wv   = tid >> 5;               // 8 waves
  const int tok0 = tile * 16;

  // ---- phase A: active mask + numeric features (threads 0..15) ----
  if (tid < 16) {
    const int tok = tok0 + tid;
    const int id  = ids[tok];
    const float v = vals[tok];
    const bool act = (id == NUM_ID) && !(v != v);
    sAct[tid] = act ? 1 : 0;
    compute_feats(act ? v : 1.0f, fmts[tok], &sFeats[tid * FP]);
  }
  __syncthreads();

  bool anyAct = false;
#pragma unroll
  for (int i = 0; i < 16; ++i) anyAct |= (sAct[i] != 0);   // block-uniform

  const int half = lane >> 4;     // WMMA fragment half (lanes 16..31)
  const int lm   = lane & 15;

  if (anyAct) {   // uniform branch: EXEC stays all-ones for WMMA
    // ---- layer 1: feats[16,96] x p_w1[96,256] -> GELU -> sH1 ----
    for (int t = 0; t < 2; ++t) {
      const int n0 = (wv + t * 8) * 16;
      const int col = n0 + lm;
      v8f acc = {0.f,0.f,0.f,0.f,0.f,0.f,0.f,0.f};
      for (int kk = 0; kk < 24; ++kk) {
        const int kb = kk * 4 + half * 2;
        v2f a; a.x = sFeats[lm * FP + kb]; a.y = sFeats[lm * FP + kb + 1];
        v2f b;
        b.x = (kb     < NF) ? pw1[kb * PI + col]       : 0.0f;
        b.y = (kb + 1 < NF) ? pw1[(kb + 1) * PI + col] : 0.0f;
        acc = __builtin_amdgcn_wmma_f32_16x16x4_f32(
            false, a, false, b, (short)0, acc, false, false);
      }
      const float bias = pb1[col];
#pragma unroll
      for (int r = 0; r < 8; ++r) {
        const int row = r + half * 8;
        const float xg = acc[r] + bias;
        sH1[row * H1S + col] = 0.5f * xg * (1.0f + erff(xg * 0.70710678118654752f));
      }
    }
    __syncthreads();
    // ---- layer 2: h1[16,256] x p_w2[256,1024] + b2 -> sNum (pre-LN) ----
    for (int t = 0; t < 8; ++t) {
      const int n0 = (wv + t * 8) * 16;
      const int col = n0 + lm;
      v8f acc = {0.f,0.f,0.f,0.f,0.f,0.f,0.f,0.f};
      for (int kk = 0; kk < 64; ++kk) {
        const int kb = kk * 4 + half * 2;
        v2f a; a.x = sH1[lm * H1S + kb]; a.y = sH1[lm * H1S + kb + 1];
        v2f b;
        b.x = pw2[kb * HID + col];
        b.y = pw2[(kb + 1) * HID + col];
        acc = __builtin_amdgcn_wmma_f32_16x16x4_f32(
            false, a, false, b, (short)0, acc, false, false);
      }
      const float bias = pb2[col];
#pragma unroll
      for (int r = 0; r < 8; ++r) {
        const int row = r + half * 8;
        sNum[row * NS + col] = acc[r] + bias;
      }
    }
  }
  __syncthreads();

  // ---- final: text gather + masked LN(pln) + add + LN(ln) ----
  // one token per wave iteration: row reductions = pure wave32 shuffles
  for (int ti = 0; ti < 2; ++ti) {
    const int local = wv * 2 + ti;
    const int tok   = tok0 + local;
    const int id    = ids[tok];
    const int pos   = tok & (SEQ - 1);
    const float4* wr = (const float4*)(Wword + (size_t)id * HID);
    const float4* pr = (const float4*)(Wpos + (size_t)pos * HID);
    const float4* tr = (const float4*)(Wtype);       // token_type row 0

    float x[32];
#pragma unroll
    for (int c = 0; c < 8; ++c) {
      const int q = c * 32 + lane;                   // float4 index 0..255
      float4 a = wr[q], b = pr[q], t = tr[q];
      x[c*4+0] = a.x + b.x + t.x;
      x[c*4+1] = a.y + b.y + t.y;
      x[c*4+2] = a.z + b.z + t.z;
      x[c*4+3] = a.w + b.w + t.w;
    }

    if (anyAct && sAct[local]) {                     // wave-uniform predicate
      float n[32];
      float s = 0.0f;
      const float* row = &sNum[local * NS];
#pragma unroll
      for (int c = 0; c < 8; ++c) {
        const int q = c * 32 + lane;
        float4 v = *(const float4*)&row[q * 4];
        n[c*4+0] = v.x; n[c*4+1] = v.y; n[c*4+2] = v.z; n[c*4+3] = v.w;
        s += v.x + v.y + v.z + v.w;
      }
      const float mu = wave_sum32(s) * (1.0f / (float)HID);
      float sq = 0.0f;
#pragma unroll
      for (int j = 0; j < 32; ++j) { const float d = n[j] - mu; sq += d * d; }
      const float var  = wave_sum32(sq) * (1.0f / (float)HID);
      const float rstd = 1.0f / sqrtf(var + LN_EPS);
      const float4* gg = (const float4*)plng;
      const float4* bb = (const float4*)plnb;
#pragma unroll
      for (int c = 0; c < 8; ++c) {
        const int q = c * 32 + lane;
        float4 g = gg[q], b = bb[q];
        x[c*4+0] += (n[c*4+0] - mu) * rstd * g.x + b.x;
        x[c*4+1] += (n[c*4+1] - mu) * rstd * g.y + b.y;
        x[c*4+2] += (n[c*4+2] - mu) * rstd * g.z + b.z;
        x[c*4+3] += (n[c*4+3] - mu) * rstd * g.w + b.w;
      }
    }

    // final LayerNorm
    float s = 0.0f;
#pragma unroll
    for (int j = 0; j < 32; ++j) s += x[j];
    const float mu = wave_sum32(s) * (1.0f / (float)HID);
    float sq = 0.0f;
#pragma unroll
    for (int j = 0; j < 32; ++j) { const float d = x[j] - mu; sq += d * d; }
    const float var  = wave_sum32(sq) * (1.0f / (float)HID);
    const float rstd = 1.0f / sqrtf(var + LN_EPS);

    const float4* gg = (const float4*)lng;
    const float4* bb = (const float4*)lnb;
    float* op = out + (size_t)tok * HID;
#pragma unroll
    for (int c = 0; c < 8; ++c) {
      const int q = c * 32 + lane;
      float4 g = gg[q], b = bb[q];
      float4 o;
      o.x = (x[c*4+0] - mu) * rstd * g.x + b.x;
      o.y = (x[c*4+1] - mu) * rstd * g.y + b.y;
      o.z = (x[c*4+2] - mu) * rstd * g.z + b.z;
      o.w = (x[c*4+3] - mu) * rstd * g.w + b.w;
      *(float4*)&op[q * 4] = o;
    }
  }
}

extern "C" void kernel_launch(void* const* d_in, const int* in_sizes, int n_in,
                              void* d_out, int out_size, void* d_ws, size_t ws_size,
                              hipStream_t stream) {
  (void)in_sizes; (void)n_in; (void)d_ws; (void)ws_size; (void)out_size;
  const int*   ids   = (const int*)  d_in[0];
  const float* vals  = (const float*)d_in[1];
  const int*   fmts  = (const int*)  d_in[2];
  const float* Wword = (const float*)d_in[3];
  const float* Wpos  = (const float*)d_in[4];
  const float* Wtype = (const float*)d_in[5];
  const float* lng   = (const float*)d_in[6];
  const float* lnb   = (const float*)d_in[7];
  const float* pw1   = (const float*)d_in[8];
  const float* pb1   = (const float*)d_in[9];
  const float* pw2   = (const float*)d_in[10];
  const float* pb2   = (const float*)d_in[11];
  const float* plng  = (const float*)d_in[12];
  const float* plnb  = (const float*)d_in[13];
  float* out = (float*)d_out;

  const size_t smem = (size_t)(16 * FP + 16 * H1S + 16 * NS) * sizeof(float)
                      + 16 * sizeof(int);
  // Raise dynamic-LDS cap (host-side config; deterministic, capture-safe).
  (void)hipFuncSetAttribute((const void*)blackhole_emb_kernel,
                            hipFuncAttributeMaxDynamicSharedMemorySize,
                            (int)smem);

  blackhole_emb_kernel<<<NTOK / 16, 256, smem, stream>>>(
      ids, vals, fmts, Wword, Wpos, Wtype, lng, lnb,
      pw1, pb1, pw2, pb2, plng, plnb, out);
}